// AtomGroupBridgeFiLM_29326036697576
// MI455X (gfx1250) — compile-verified
//
#include <hip/hip_runtime.h>
#include <math.h>

// ---------------- problem constants ----------------
#define NA_C    500000
#define NINC_C  500000
#define GM_C    100000
#define AD_C    180     // ATOM_DIM
#define CD_C    64      // COND_DIM
#define H_C     80      // S2S_H

// output layout (floats)
#define OUT_XG_OFF   ((size_t)NA_C * AD_C)            // 90,000,000
#define OUT_TID_OFF  (OUT_XG_OFF + (size_t)GM_C*263)  // 116,300,000

// workspace layout (float indices)
#define OFF_R1       ((size_t)0)            // GM*80
#define OFF_R2       ((size_t)8000000)      // GM*80
#define OFF_DEN1     ((size_t)16000000)     // GM
#define OFF_DEN2     ((size_t)16100000)     // GM
#define OFF_CNT      ((size_t)16200000)     // GM
#define OFF_EMAX1    ((size_t)16300000)     // GM (uint keys)
#define OFF_EMAX2    ((size_t)16400000)     // GM (uint keys)
#define OFF_CONDG    ((size_t)16500000)     // GM*64 (sum -> mean in place)
#define ZERO_FLOATS  ((size_t)22900000)     // contiguous zero-init region
#define OFF_XAPROJ   ((size_t)22900000)     // NA*80
#define OFF_Q2       ((size_t)62900000)     // GM*80
#define OFF_E        ((size_t)70900000)     // NINC
#define OFF_H1       ((size_t)71400000)     // 80
#define OFF_C1       ((size_t)71400080)     // 80
#define OFF_CONST2   ((size_t)71400160)     // 320

typedef float v2f __attribute__((ext_vector_type(2)));
typedef float v8f __attribute__((ext_vector_type(8)));

__device__ __forceinline__ v8f wmma4(v2f a, v2f b, v8f c) {
  // D = A(16x4,f32) * B(4x16,f32) + C(16x16,f32)   (exact f32 tensor op)
  return __builtin_amdgcn_wmma_f32_16x16x4_f32(false, a, false, b, (short)0, c, false, false);
}
__device__ __forceinline__ v8f vfill(float x) {
  v8f v;
#pragma unroll
  for (int i = 0; i < 8; ++i) v[i] = x;
  return v;
}
__device__ __forceinline__ float sigm(float x) { return 1.0f / (1.0f + expf(-x)); }

// monotonic float<->uint key for atomicMax-based segment max
__device__ __forceinline__ unsigned fkey(float f) {
  unsigned u = __float_as_uint(f);
  return (u & 0x80000000u) ? ~u : (u | 0x80000000u);
}
__device__ __forceinline__ float funkey(unsigned k) {
  unsigned u = (k & 0x80000000u) ? (k & 0x7fffffffu) : ~k;
  return __uint_as_float(u);
}

// ---------------- step-1 constants: h1, c1, const2 ----------------
__global__ void consts_kernel(const float* __restrict__ b_ih, const float* __restrict__ b_hh,
                              const float* __restrict__ w_ih, const float* __restrict__ w_hh,
                              float* __restrict__ h1, float* __restrict__ c1,
                              float* __restrict__ const2) {
  __shared__ float g1[320];
  __shared__ float h1s[80];
  int t = threadIdx.x; // 320 threads
  float gv = b_ih[t] + b_hh[t];
  g1[t] = gv;
  __syncthreads();
  if (t < 80) {
    float iv = g1[t], gg = g1[160 + t], ov = g1[240 + t];
    float c = sigm(iv) * tanhf(gg);   // f-gate * c0(=0) drops
    float h = sigm(ov) * tanhf(c);
    c1[t] = c; h1[t] = h; h1s[t] = h;
  }
  __syncthreads();
  float acc = gv;
#pragma unroll 4
  for (int j = 0; j < 80; ++j)
    acc += h1s[j] * (w_hh[(size_t)t * 80 + j] + w_ih[(size_t)t * 160 + j]);
  const2[t] = acc;
}

// ---------------- xa_proj = x_atom @ a2g_w.T + b  (+ fused x_atom copy) ----------------
__global__ void __launch_bounds__(64) xaproj_kernel(const float* __restrict__ xa,
                                                    const float* __restrict__ W,
                                                    const float* __restrict__ bias,
                                                    float* __restrict__ xaproj,
                                                    float* __restrict__ out_copy) {
  int wave = threadIdx.x >> 5, lane = threadIdx.x & 31;
  int tile = blockIdx.x * 2 + wave;            // exact: 31250 tiles
  int cn = lane & 15, kh = lane >> 4;
  const float* arow = xa + (size_t)(tile * 16 + cn) * AD_C;
  float* crow = out_copy + (size_t)(tile * 16 + cn) * AD_C;

  v8f acc[5];
#pragma unroll
  for (int nt = 0; nt < 5; ++nt) acc[nt] = vfill(bias[nt * 16 + cn]);

  for (int ks = 0; ks < 45; ++ks) {
    int ka = 4 * ks + 2 * kh;
    v2f a; a.x = arow[ka]; a.y = arow[ka + 1];
    crow[ka] = a.x; crow[ka + 1] = a.y;        // fused passthrough copy
#pragma unroll
    for (int nt = 0; nt < 5; ++nt) {
      const float* wr = W + (size_t)(nt * 16 + cn) * AD_C + ka;
      v2f b; b.x = wr[0]; b.y = wr[1];
      acc[nt] = wmma4(a, b, acc[nt]);
    }
  }
#pragma unroll
  for (int nt = 0; nt < 5; ++nt)
#pragma unroll
    for (int r = 0; r < 8; ++r)
      xaproj[(size_t)(tile * 16 + r + 8 * kh) * H_C + nt * 16 + cn] = acc[nt][r];
}

// ---------------- counts + cond segment sum ----------------
__global__ void count_cond_kernel(const int* __restrict__ atom_idx, const int* __restrict__ group_idx,
                                  const float* __restrict__ cond_atom,
                                  float* __restrict__ cnt, float* __restrict__ cond_sum) {
  int i = blockIdx.x * blockDim.x + threadIdx.x;
  if (i >= NINC_C) return;
  int a = atom_idx[i], g = group_idx[i];
  atomicAdd(&cnt[g], 1.0f);
  const float* cr = cond_atom + (size_t)a * CD_C;
  float* dst = cond_sum + (size_t)g * CD_C;
#pragma unroll 4
  for (int c = 0; c < CD_C; ++c) atomicAdd(&dst[c], cr[c]);
}

__global__ void cond_finalize_kernel(float* __restrict__ cond, const float* __restrict__ cnt,
                                     float* __restrict__ out) {
  int t = blockIdx.x * blockDim.x + threadIdx.x;
  if (t >= GM_C * CD_C) return;
  int g = t >> 6, c = t & 63;
  float v = cond[t] / fmaxf(cnt[g], 1.0f);
  cond[t] = v;                                           // cond_g in place
  out[OUT_XG_OFF + (size_t)g * 263 + 199 + c] = v;       // concat tail
}

// ---------------- attention pass 1: e + segment max ----------------
__global__ void e_pass_kernel(const float* __restrict__ xaproj, const int* __restrict__ atom_idx,
                              const int* __restrict__ group_idx,
                              const float* __restrict__ qconst, const float* __restrict__ qarr,
                              float* __restrict__ e, unsigned* __restrict__ emaxk) {
  int i = blockIdx.x * blockDim.x + threadIdx.x;
  if (i >= NINC_C) return;
  int a = atom_idx[i], g = group_idx[i];
  const float4* x4 = (const float4*)(xaproj + (size_t)a * H_C);
  const float* q = qconst ? qconst : (qarr + (size_t)g * H_C);
  const float4* q4 = (const float4*)q;
  float s = 0.f;
#pragma unroll
  for (int k = 0; k < 20; ++k) {
    float4 xv = x4[k], qv = q4[k];
    s += xv.x * qv.x + xv.y * qv.y + xv.z * qv.z + xv.w * qv.w;
  }
  e[i] = s;
  atomicMax(&emaxk[g], fkey(s));
}

// ---------------- attention pass 2: exp, denom, r_raw ----------------
__global__ void adr_kernel(const float* __restrict__ xaproj, const int* __restrict__ atom_idx,
                           const int* __restrict__ group_idx, const float* __restrict__ e,
                           const unsigned* __restrict__ emaxk,
                           float* __restrict__ denom, float* __restrict__ r_raw) {
  int i = blockIdx.x * blockDim.x + threadIdx.x;
  if (i >= NINC_C) return;
  int a = atom_idx[i], g = group_idx[i];
  float w = expf(e[i] - funkey(emaxk[g]));
  atomicAdd(&denom[g], w);
  const float* x = xaproj + (size_t)a * H_C;
  float* dst = r_raw + (size_t)g * H_C;
#pragma unroll 4
  for (int k = 0; k < H_C; ++k) atomicAdd(&dst[k], w * x[k]);
}

// ---------------- LSTM step 2: gates = (r1/den1) @ w_ih_r.T + const2 -> q2 ----------------
__global__ void __launch_bounds__(64) lstm2_kernel(const float* __restrict__ r1,
                                                   const float* __restrict__ den1,
                                                   const float* __restrict__ w_ih,
                                                   const float* __restrict__ const2,
                                                   const float* __restrict__ c1,
                                                   float* __restrict__ q2) {
  __shared__ float gAll[2][16 * 320];
  int wave = threadIdx.x >> 5, lane = threadIdx.x & 31;
  int tile = blockIdx.x * 2 + wave;            // exact: 6250 tiles
  int g0 = tile * 16, cn = lane & 15, kh = lane >> 4;
  float* ldsG = gAll[wave];

  float inv = 1.0f / (den1[g0 + cn] + 1e-16f);
  const float* rrow = r1 + (size_t)(g0 + cn) * H_C;
  v2f af[20];
#pragma unroll
  for (int ks = 0; ks < 20; ++ks) {
    int ka = 4 * ks + 2 * kh;
    af[ks].x = rrow[ka] * inv; af[ks].y = rrow[ka + 1] * inv;
  }
  for (int nt = 0; nt < 20; ++nt) {
    int n = nt * 16 + cn;
    v8f acc = vfill(const2[n]);
#pragma unroll
    for (int ks = 0; ks < 20; ++ks) {
      const float* wr = w_ih + (size_t)n * 160 + 80 + 4 * ks + 2 * kh;
      v2f b; b.x = wr[0]; b.y = wr[1];
      acc = wmma4(af[ks], b, acc);
    }
#pragma unroll
    for (int r = 0; r < 8; ++r) ldsG[(r + 8 * kh) * 320 + n] = acc[r];
  }
  __syncthreads();
  for (int idx = lane; idx < 16 * H_C; idx += 32) {
    int m = idx / H_C, j = idx - m * H_C;
    const float* gm = ldsG + m * 320;
    float c2 = sigm(gm[80 + j]) * c1[j] + sigm(gm[j]) * tanhf(gm[160 + j]);
    float h2 = sigm(gm[240 + j]) * tanhf(c2);
    q2[(size_t)(g0 + m) * H_C + j] = h2;
  }
}

// ---------------- fused FiLM MLP half: hidden=relu(cond@W1.T+b1); out(=gamma/beta) applied to bufM ----------------
__device__ __forceinline__ void film_half(const float* __restrict__ crow,
                                          const float* __restrict__ w1, const float* __restrict__ b1,
                                          const float* __restrict__ w2, const float* __restrict__ b2,
                                          float* bufA, float* bufM, int cn, int kh, bool isGamma) {
  // hidden: (16x64)@(64x202) -> bufA (16x208, zero-padded)
  for (int nt = 0; nt < 13; ++nt) {
    int n = nt * 16 + cn; int nc = n < 202 ? n : 201; float nm = n < 202 ? 1.f : 0.f;
    v8f acc = vfill(b1[nc] * nm);
    const float* wr = w1 + (size_t)nc * 64;
    for (int ks = 0; ks < 16; ++ks) {
      int ka = 4 * ks + 2 * kh;
      v2f a; a.x = crow[ka]; a.y = crow[ka + 1];
      v2f b; b.x = wr[ka] * nm; b.y = wr[ka + 1] * nm;
      acc = wmma4(a, b, acc);
    }
#pragma unroll
    for (int r = 0; r < 8; ++r) bufA[(r + 8 * kh) * 208 + n] = fmaxf(acc[r], 0.f);
  }
  __syncthreads();
  // outer: (16x208)@(208x199) with zero-padded K; apply to bufM
  for (int nt = 0; nt < 13; ++nt) {
    int n = nt * 16 + cn; int nc = n < 199 ? n : 198; float nm = n < 199 ? 1.f : 0.f;
    v8f acc = vfill(b2[nc] * nm);
    const float* wr = w2 + (size_t)nc * 202;
    for (int ks = 0; ks < 52; ++ks) {
      int ka = 4 * ks + 2 * kh; int kc = ka < 202 ? ka : 200;  // A is 0 in pad cols
      v2f a; a.x = bufA[cn * 208 + ka]; a.y = bufA[cn * 208 + ka + 1];
      v2f b; b.x = wr[kc] * nm; b.y = wr[kc + 1] * nm;
      acc = wmma4(a, b, acc);
    }
#pragma unroll
    for (int r = 0; r < 8; ++r) {
      int idx = (r + 8 * kh) * 208 + n;
      if (isGamma) bufM[idx] *= acc[r]; else bufM[idx] += acc[r];
    }
  }
  __syncthreads();
}

// ---------------- final fused: xg_feat + merge + FiLM -> out ----------------
__global__ void __launch_bounds__(64) final_kernel(
    const float* __restrict__ x_group, const float* __restrict__ g_proj_w, const float* __restrict__ g_proj_b,
    const float* __restrict__ q2, const float* __restrict__ r2, const float* __restrict__ den2,
    const float* __restrict__ cnt, const float* __restrict__ merge_w, const float* __restrict__ merge_b,
    const float* __restrict__ cond_g,
    const float* __restrict__ fg1_w, const float* __restrict__ fg1_b,
    const float* __restrict__ fg2_w, const float* __restrict__ fg2_b,
    const float* __restrict__ fb1_w, const float* __restrict__ fb1_b,
    const float* __restrict__ fb2_w, const float* __restrict__ fb2_b,
    float* __restrict__ out) {
  __shared__ float sA[2][16 * 208];
  __shared__ float sM[2][16 * 208];
  int wave = threadIdx.x >> 5, lane = threadIdx.x & 31;
  int tile = blockIdx.x * 2 + wave;           // exact: 6250 tiles
  int g0 = tile * 16, cn = lane & 15, kh = lane >> 4;
  float* bufA = sA[wave]; float* bufM = sM[wave];

  float mask = cnt[g0 + cn] > 0.f ? 1.f : 0.f;
  float inv = mask / (den2[g0 + cn] + 1e-16f);

  // xg_feat = Xg40 @ g_proj_w.T + b  -> bufA cols [0,80)
  const float* xgrow = x_group + (size_t)(g0 + cn) * 160;
  for (int nt = 0; nt < 5; ++nt) {
    v8f acc = vfill(g_proj_b[nt * 16 + cn]);
    for (int ks = 0; ks < 10; ++ks) {
      int ka = 4 * ks + 2 * kh;
      v2f a; a.x = xgrow[ka]; a.y = xgrow[ka + 1];
      const float* wr = g_proj_w + (size_t)(nt * 16 + cn) * 40 + ka;
      v2f b; b.x = wr[0]; b.y = wr[1];
      acc = wmma4(a, b, acc);
    }
#pragma unroll
    for (int r = 0; r < 8; ++r) bufA[(r + 8 * kh) * 208 + nt * 16 + cn] = acc[r];
  }
  __syncthreads();

  // merge: concat(xg_feat, mask*q2, mask*r2) @ merge_w.T + b  -> bufM
  const float* q2row = q2 + (size_t)(g0 + cn) * H_C;
  const float* r2row = r2 + (size_t)(g0 + cn) * H_C;
  for (int nt = 0; nt < 13; ++nt) {
    int n = nt * 16 + cn; int nc = n < 199 ? n : 198; float nm = n < 199 ? 1.f : 0.f;
    v8f acc = vfill(merge_b[nc] * nm);
    for (int ks = 0; ks < 60; ++ks) {
      int k0 = 4 * ks; int ka = k0 + 2 * kh;
      v2f a;
      if (k0 < 80)        { a.x = bufA[cn * 208 + ka];        a.y = bufA[cn * 208 + ka + 1]; }
      else if (k0 < 160)  { int kk = ka - 80;  a.x = q2row[kk] * mask; a.y = q2row[kk + 1] * mask; }
      else                { int kk = ka - 160; a.x = r2row[kk] * inv;  a.y = r2row[kk + 1] * inv; }
      const float* wr = merge_w + (size_t)nc * 240 + ka;
      v2f b; b.x = wr[0] * nm; b.y = wr[1] * nm;
      acc = wmma4(a, b, acc);
    }
#pragma unroll
    for (int r = 0; r < 8; ++r) bufM[(r + 8 * kh) * 208 + n] = acc[r];
  }
  __syncthreads();

  // FiLM: gamma multiplies, beta adds
  const float* crow = cond_g + (size_t)(g0 + cn) * CD_C;
  film_half(crow, fg1_w, fg1_b, fg2_w, fg2_b, bufA, bufM, cn, kh, true);
  film_half(crow, fb1_w, fb1_b, fb2_w, fb2_b, bufA, bufM, cn, kh, false);

  // store xg (cols 0..198); cols 199..262 already written by cond_finalize
  for (int idx = lane; idx < 16 * 199; idx += 32) {
    int m = idx / 199, n = idx - m * 199;
    out[OUT_XG_OFF + (size_t)(g0 + m) * 263 + n] = bufM[m * 208 + n];
  }
}

// ---------------- argmax over first 40 cols of x_group ----------------
__global__ void argmax_kernel(const float* __restrict__ x_group, float* __restrict__ out) {
  int g = blockIdx.x * blockDim.x + threadIdx.x;
  if (g >= GM_C) return;
  const float* row = x_group + (size_t)g * 160;
  float best = row[0]; int bi = 0;
#pragma unroll 4
  for (int c = 1; c < 40; ++c) { float v = row[c]; if (v > best) { best = v; bi = c; } }
  out[OUT_TID_OFF + g] = (float)bi;
}

// ---------------- launcher ----------------
extern "C" void kernel_launch(void* const* d_in, const int* in_sizes, int n_in,
                              void* d_out, int out_size, void* d_ws, size_t ws_size,
                              hipStream_t stream) {
  const float* x_atom    = (const float*)d_in[0];
  const int*   atom_idx  = (const int*)d_in[1];
  const float* x_group   = (const float*)d_in[2];
  const int*   group_idx = (const int*)d_in[3];
  // d_in[4] edge_index_group: unused by reference
  const float* cond_atom = (const float*)d_in[5];
  const float* g_proj_w  = (const float*)d_in[6];
  const float* g_proj_b  = (const float*)d_in[7];
  const float* a2g_w     = (const float*)d_in[8];
  const float* a2g_b     = (const float*)d_in[9];
  const float* merge_w   = (const float*)d_in[10];
  const float* merge_b   = (const float*)d_in[11];
  const float* fg1_w     = (const float*)d_in[12];
  const float* fg1_b     = (const float*)d_in[13];
  const float* fg2_w     = (const float*)d_in[14];
  const float* fg2_b     = (const float*)d_in[15];
  const float* fb1_w     = (const float*)d_in[16];
  const float* fb1_b     = (const float*)d_in[17];
  const float* fb2_w     = (const float*)d_in[18];
  const float* fb2_b     = (const float*)d_in[19];
  const float* lstm_w_ih = (const float*)d_in[20];
  const float* lstm_w_hh = (const float*)d_in[21];
  const float* lstm_b_ih = (const float*)d_in[22];
  const float* lstm_b_hh = (const float*)d_in[23];

  float* out = (float*)d_out;
  float* ws  = (float*)d_ws;
  unsigned* emax1 = (unsigned*)(ws + OFF_EMAX1);
  unsigned* emax2 = (unsigned*)(ws + OFF_EMAX2);

  hipMemsetAsync(ws, 0, ZERO_FLOATS * sizeof(float), stream);

  consts_kernel<<<1, 320, 0, stream>>>(lstm_b_ih, lstm_b_hh, lstm_w_ih, lstm_w_hh,
                                       ws + OFF_H1, ws + OFF_C1, ws + OFF_CONST2);

  xaproj_kernel<<<15625, 64, 0, stream>>>(x_atom, a2g_w, a2g_b, ws + OFF_XAPROJ, out);

  count_cond_kernel<<<(NINC_C + 255) / 256, 256, 0, stream>>>(atom_idx, group_idx, cond_atom,
                                                              ws + OFF_CNT, ws + OFF_CONDG);
  cond_finalize_kernel<<<(GM_C * CD_C) / 256, 256, 0, stream>>>(ws + OFF_CONDG, ws + OFF_CNT, out);

  // set2set step 1 (query = constant h1)
  e_pass_kernel<<<(NINC_C + 255) / 256, 256, 0, stream>>>(ws + OFF_XAPROJ, atom_idx, group_idx,
                                                          ws + OFF_H1, nullptr, ws + OFF_E, emax1);
  adr_kernel<<<(NINC_C + 255) / 256, 256, 0, stream>>>(ws + OFF_XAPROJ, atom_idx, group_idx,
                                                       ws + OFF_E, emax1, ws + OFF_DEN1, ws + OFF_R1);

  lstm2_kernel<<<3125, 64, 0, stream>>>(ws + OFF_R1, ws + OFF_DEN1, lstm_w_ih,
                                        ws + OFF_CONST2, ws + OFF_C1, ws + OFF_Q2);

  // set2set step 2 (query = q2 per group)
  e_pass_kernel<<<(NINC_C + 255) / 256, 256, 0, stream>>>(ws + OFF_XAPROJ, atom_idx, group_idx,
                                                          nullptr, ws + OFF_Q2, ws + OFF_E, emax2);
  adr_kernel<<<(NINC_C + 255) / 256, 256, 0, stream>>>(ws + OFF_XAPROJ, atom_idx, group_idx,
                                                       ws + OFF_E, emax2, ws + OFF_DEN2, ws + OFF_R2);

  final_kernel<<<3125, 64, 0, stream>>>(x_group, g_proj_w, g_proj_b,
                                        ws + OFF_Q2, ws + OFF_R2, ws + OFF_DEN2, ws + OFF_CNT,
                                        merge_w, merge_b, ws + OFF_CONDG,
                                        fg1_w, fg1_b, fg2_w, fg2_b,
                                        fb1_w, fb1_b, fb2_w, fb2_b, out);

  argmax_kernel<<<(GM_C + 255) / 256, 256, 0, stream>>>(x_group, out);
}